// Iwt2d_7928509629348
// MI455X (gfx1250) — compile-verified
//
#include <hip/hip_runtime.h>

// IWT 2D (inverse Haar recombine): x (16,256,128,128) f32 -> out (16,64,256,256) f32
// Pure bandwidth-bound streaming kernel: 537 MB total traffic, ~23 us floor @ 23.3 TB/s.
// Strategy: b128 loads/stores, non-temporal hints (working set > 192 MB L2),
// global_prefetch_b8 a few rows ahead.

typedef float v4f __attribute__((ext_vector_type(4)));

__global__ __launch_bounds__(256) void iwt2d_kernel(const float* __restrict__ x,
                                                    float* __restrict__ out) {
    constexpr unsigned CH = 64, Hs = 128, Ws = 128;
    constexpr unsigned IN_PLANE  = Hs * Ws;          // 16384 floats per (b,c) plane
    constexpr unsigned IN_GROUP  = CH * IN_PLANE;    // stride between LL/LH/HL/HH groups
    constexpr unsigned IN_BATCH  = 4u * IN_GROUP;    // 4,194,304
    constexpr unsigned OUT_PLANE = (2u * Hs) * (2u * Ws);  // 65536
    constexpr unsigned OUT_BATCH = CH * OUT_PLANE;   // 4,194,304
    constexpr unsigned OW = 2u * Ws;                 // 256

    unsigned gid = blockIdx.x * 256u + threadIdx.x;
    unsigned w4 = gid & 31u;            // which float4 chunk along W (W/4 = 32)
    unsigned h  = (gid >> 5) & 127u;
    unsigned c  = (gid >> 12) & 63u;
    unsigned b  = gid >> 18;

    unsigned w0   = w4 << 2;
    unsigned base = b * IN_BATCH + c * IN_PLANE + h * Ws + w0;

    const float* pLL = x + base;
    const float* pLH = pLL + IN_GROUP;
    const float* pHL = pLL + 2u * IN_GROUP;
    const float* pHH = pLL + 3u * IN_GROUP;

    // One speculative prefetch per 128 B per stream, ~16 rows (8 KB) ahead.
    // Speculative prefetches past the allocation are silently dropped.
    if ((gid & 7u) == 0u) {
        constexpr unsigned PF = 16u * Ws;
        __builtin_prefetch(pLL + PF, 0, 0);
        __builtin_prefetch(pLH + PF, 0, 0);
        __builtin_prefetch(pHL + PF, 0, 0);
        __builtin_prefetch(pHH + PF, 0, 0);
    }

    // Non-temporal b128 loads: each element is consumed exactly once.
    v4f ll = __builtin_nontemporal_load((const v4f*)pLL);
    v4f lh = __builtin_nontemporal_load((const v4f*)pLH);
    v4f hl = __builtin_nontemporal_load((const v4f*)pHL);
    v4f hh = __builtin_nontemporal_load((const v4f*)pHH);

    // Butterfly
    v4f s01 = ll + lh;   // LL+LH
    v4f d01 = ll - lh;   // LL-LH
    v4f s23 = hl + hh;   // HL+HH
    v4f d23 = hl - hh;   // HL-HH

    v4f av = (s01 + s23) * 0.5f;  // a -> (2h  , 2w  )
    v4f bv = (s01 - s23) * 0.5f;  // b -> (2h  , 2w+1)
    v4f cv = (d01 + d23) * 0.5f;  // c -> (2h+1, 2w  )
    v4f dv = (d01 - d23) * 0.5f;  // d -> (2h+1, 2w+1)

    // Interleave 4 positions into two 8-float (two b128) runs per output row.
    v4f t0 = {av.x, bv.x, av.y, bv.y};
    v4f t1 = {av.z, bv.z, av.w, bv.w};
    v4f u0 = {cv.x, dv.x, cv.y, dv.y};
    v4f u1 = {cv.z, dv.z, cv.w, dv.w};

    unsigned ob = b * OUT_BATCH + c * OUT_PLANE + (2u * h) * OW + 2u * w0;
    float* otop = out + ob;
    float* obot = otop + OW;

    // Non-temporal b128 stores: written once, never re-read; don't pollute L2.
    __builtin_nontemporal_store(t0, (v4f*)otop);
    __builtin_nontemporal_store(t1, (v4f*)(otop + 4));
    __builtin_nontemporal_store(u0, (v4f*)obot);
    __builtin_nontemporal_store(u1, (v4f*)(obot + 4));
}

extern "C" void kernel_launch(void* const* d_in, const int* in_sizes, int n_in,
                              void* d_out, int out_size, void* d_ws, size_t ws_size,
                              hipStream_t stream) {
    (void)in_sizes; (void)n_in; (void)d_ws; (void)ws_size; (void)out_size;
    const float* x = (const float*)d_in[0];
    float* out = (float*)d_out;

    // 16 * 64 * 128 * 32 threads = 4,194,304 -> 16384 blocks of 256 (8 wave32 per block)
    constexpr unsigned total_threads = 16u * 64u * 128u * 32u;
    dim3 grid(total_threads / 256u);
    dim3 block(256u);
    iwt2d_kernel<<<grid, block, 0, stream>>>(x, out);
}